// GNN2_6184752906610
// MI455X (gfx1250) — compile-verified
//
#include <hip/hip_runtime.h>
#include <hip/hip_bf16.h>

typedef __attribute__((ext_vector_type(2))) float v2f;
typedef __attribute__((ext_vector_type(8))) float v8f;

#define GCN_N_NODES  100000
#define GCN_N_EDGES  3200000
#define GCN_N_GRAPHS 1000

// ---------------- degree / normalization ----------------

__global__ void gcn_deg_init(float* __restrict__ deg, int n) {
    int i = blockIdx.x * blockDim.x + threadIdx.x;
    if (i < n) deg[i] = 1.0f;   // self-loop contributes 1 to every node's degree
}

__global__ void gcn_deg_accum(const int* __restrict__ col, float* __restrict__ deg, int e) {
    int i = blockIdx.x * blockDim.x + threadIdx.x;
    if (i < e) atomicAdd(&deg[col[i]], 1.0f);
}

__global__ void gcn_deg_to_dinv(float* __restrict__ deg, int n) {
    int i = blockIdx.x * blockDim.x + threadIdx.x;
    if (i < n) deg[i] = rsqrtf(deg[i]);   // deg >= 1 always (self-loops)
}

// ---------------- fp32 WMMA GEMM:  out[m,n] = dinv[m] * sum_k X[m,k]*W[k,n] --------------
// One wave (32 lanes) computes one 16x16 tile of the output via V_WMMA_F32_16X16X4_F32.
// A (16x4 f32): lane = M row, half selects K pair: v0 = K+2h, v1 = K+2h+1.
// B (4x16 f32): lane = N col, half selects K pair (mirror of A).
// D (16x16 f32, 8 VGPRs): VGPR j -> M = j + 8*half, N = lane&15.

template <int K, int F>
__global__ __launch_bounds__(32)
void gcn_gemm_wmma_scaled(const float* __restrict__ X, const float* __restrict__ W,
                          const float* __restrict__ dinv, float* __restrict__ out) {
    const int mt   = blockIdx.x;
    const int nt   = blockIdx.y;
    const int lane = threadIdx.x & 31;
    const int half = lane >> 4;
    const int l    = lane & 15;
    const int m0   = mt * 16;
    const int n0   = nt * 16;

    const float* __restrict__ xrow = X + (size_t)(m0 + l) * K;

    v8f acc = {};
#pragma unroll 4
    for (int k = 0; k < K; k += 4) {
        v2f a, b;
        a[0] = xrow[k + 2 * half];
        a[1] = xrow[k + 2 * half + 1];
        b[0] = W[(size_t)(k + 2 * half)     * F + n0 + l];
        b[1] = W[(size_t)(k + 2 * half + 1) * F + n0 + l];
        acc = __builtin_amdgcn_wmma_f32_16x16x4_f32(
            /*neg_a=*/false, a, /*neg_b=*/false, b,
            /*c_mod=*/(short)0, acc, /*reuse_a=*/false, /*reuse_b=*/false);
    }

#pragma unroll
    for (int j = 0; j < 8; ++j) {
        int m = m0 + j + 8 * half;
        out[(size_t)m * F + n0 + l] = acc[j] * dinv[m];   // pre-scale by dinv[row]
    }
}

// ---------------- edge scatter:  agg[c, f] += lin_s[r, f]  ----------------
// One thread per (edge, feature); feature = fastest index -> coalesced gather
// from lin_s and consecutive-address atomics into agg (L2-resident, 25.6 MB max).

template <int F>
__global__ void gcn_edge_scatter(const int* __restrict__ row, const int* __restrict__ col,
                                 const float* __restrict__ lin_s, float* __restrict__ agg,
                                 int e) {
    int t = blockIdx.x * blockDim.x + threadIdx.x;
    if (t >= e * F) return;
    int ei = t / F;
    int f  = t - ei * F;
    int r  = row[ei];
    int c  = col[ei];
    atomicAdd(&agg[(size_t)c * F + f], lin_s[(size_t)r * F + f]);
}

// ---------------- finalize:  h = relu(dinv[c]*(agg + lin_s) + b) ----------------
// (self-loop term lin[c]*dinv[c]^2 == dinv[c]*lin_s[c], folded in here)

template <int F>
__global__ void gcn_finalize(const float* __restrict__ agg, const float* __restrict__ lin_s,
                             const float* __restrict__ dinv, const float* __restrict__ bias,
                             float* __restrict__ h, int n) {
    int t = blockIdx.x * blockDim.x + threadIdx.x;
    if (t >= n * F) return;
    int i = t / F;
    int f = t - i * F;
    float v = dinv[i] * (agg[t] + lin_s[t]) + bias[f];
    h[t] = v > 0.0f ? v : 0.0f;
}

// ---------------- mean pool + linear head ----------------

__global__ void gcn_pool_accum(const int* __restrict__ batch, const float* __restrict__ h,
                               float* __restrict__ pooled, float* __restrict__ cnt, int n) {
    int t = blockIdx.x * blockDim.x + threadIdx.x;
    if (t >= n * 64) return;
    int i = t >> 6;
    int f = t & 63;
    int g = batch[i];
    atomicAdd(&pooled[(size_t)g * 64 + f], h[t]);
    if (f == 0) atomicAdd(&cnt[g], 1.0f);
}

__global__ void gcn_head(const float* __restrict__ pooled, const float* __restrict__ cnt,
                         const float* __restrict__ Wl, const float* __restrict__ bl,
                         float* __restrict__ out, int g) {
    int i = blockIdx.x * blockDim.x + threadIdx.x;
    if (i >= g) return;
    float inv = 1.0f / fmaxf(cnt[i], 1.0f);
    float s = 0.0f;
#pragma unroll
    for (int f = 0; f < 64; ++f) s += pooled[(size_t)i * 64 + f] * Wl[f];
    out[i] = s * inv + bl[0];
}

// ---------------- launch ----------------

extern "C" void kernel_launch(void* const* d_in, const int* in_sizes, int n_in,
                              void* d_out, int out_size, void* d_ws, size_t ws_size,
                              hipStream_t stream) {
    const float* x   = (const float*)d_in[0];
    const int*   ei  = (const int*)d_in[1];     // [2, E]: row = ei, col = ei + E
    const int*   bat = (const int*)d_in[2];
    const float* W1  = (const float*)d_in[3];
    const float* b1  = (const float*)d_in[4];
    const float* W2  = (const float*)d_in[5];
    const float* b2  = (const float*)d_in[6];
    const float* W3  = (const float*)d_in[7];
    const float* b3  = (const float*)d_in[8];
    const float* Wl  = (const float*)d_in[9];
    const float* bl  = (const float*)d_in[10];
    float* out = (float*)d_out;

    const int N = GCN_N_NODES, E = GCN_N_EDGES, G = GCN_N_GRAPHS;
    const int* row = ei;
    const int* col = ei + E;

    // workspace layout (floats)
    float* dinv   = (float*)d_ws;                   // N
    float* lin_s  = dinv + N;                       // N*64
    float* agg    = lin_s + (size_t)N * 64;         // N*64
    float* h      = agg   + (size_t)N * 64;         // N*64
    float* pooled = h     + (size_t)N * 64;         // G*64
    float* cnt    = pooled + (size_t)G * 64;        // G

    const int TB = 256;
    auto cdiv = [](long long a, long long b) { return (unsigned)((a + b - 1) / b); };

    // 1) dinv = rsqrt(deg) with self-loops
    gcn_deg_init<<<cdiv(N, TB), TB, 0, stream>>>(dinv, N);
    gcn_deg_accum<<<cdiv(E, TB), TB, 0, stream>>>(col, dinv, E);
    gcn_deg_to_dinv<<<cdiv(N, TB), TB, 0, stream>>>(dinv, N);

    // ---- layer 1: 128 -> 16 ----
    gcn_gemm_wmma_scaled<128, 16><<<dim3(N / 16, 1), 32, 0, stream>>>(x, W1, dinv, lin_s);
    hipMemsetAsync(agg, 0, (size_t)N * 16 * sizeof(float), stream);
    gcn_edge_scatter<16><<<cdiv((long long)E * 16, TB), TB, 0, stream>>>(row, col, lin_s, agg, E);
    gcn_finalize<16><<<cdiv((long long)N * 16, TB), TB, 0, stream>>>(agg, lin_s, dinv, b1, h, N);

    // ---- layer 2: 16 -> 32 ----
    gcn_gemm_wmma_scaled<16, 32><<<dim3(N / 16, 2), 32, 0, stream>>>(h, W2, dinv, lin_s);
    hipMemsetAsync(agg, 0, (size_t)N * 32 * sizeof(float), stream);
    gcn_edge_scatter<32><<<cdiv((long long)E * 32, TB), TB, 0, stream>>>(row, col, lin_s, agg, E);
    gcn_finalize<32><<<cdiv((long long)N * 32, TB), TB, 0, stream>>>(agg, lin_s, dinv, b2, h, N);

    // ---- layer 3: 32 -> 64 ----
    gcn_gemm_wmma_scaled<32, 64><<<dim3(N / 16, 4), 32, 0, stream>>>(h, W3, dinv, lin_s);
    hipMemsetAsync(agg, 0, (size_t)N * 64 * sizeof(float), stream);
    gcn_edge_scatter<64><<<cdiv((long long)E * 64, TB), TB, 0, stream>>>(row, col, lin_s, agg, E);
    gcn_finalize<64><<<cdiv((long long)N * 64, TB), TB, 0, stream>>>(agg, lin_s, dinv, b3, h, N);

    // ---- mean pool + head ----
    hipMemsetAsync(pooled, 0, ((size_t)G * 64 + G) * sizeof(float), stream);
    gcn_pool_accum<<<cdiv((long long)N * 64, TB), TB, 0, stream>>>(bat, h, pooled, cnt, N);
    gcn_head<<<cdiv(G, TB), TB, 0, stream>>>(pooled, cnt, Wl, bl, out, G);
}